// PureDiv32_62380105007300
// MI455X (gfx1250) — compile-verified
//
#include <hip/hip_runtime.h>

typedef __attribute__((ext_vector_type(2))) float v2f;
typedef __attribute__((ext_vector_type(8))) float v8f;

#define WAVES  8
#define HSTR   68   // padded stride for 64-wide rows (bank-conflict-free)
#define PSTR   36   // padded stride for 32-wide rows
#define WU_OFF 0
#define WG_OFF (32 * HSTR)                  // 2176
#define WD_OFF (2 * 32 * HSTR)              // 4352
#define BU_OFF (2 * 32 * HSTR + 64 * PSTR)  // 6656
#define WBUF_N (BU_OFF + 32)                // 6688 floats per weight buffer

__device__ __forceinline__ void load_weights(float* wb,
                                             const float* __restrict__ gWu,
                                             const float* __restrict__ gbu,
                                             const float* __restrict__ gWg,
                                             const float* __restrict__ gWd,
                                             int s, int tid) {
  const float4* u4 = (const float4*)(gWu + (size_t)s * 2048);
  const float4* g4 = (const float4*)(gWg + (size_t)s * 2048);
  const float4* d4 = (const float4*)(gWd + (size_t)s * 2048);
  #pragma unroll
  for (int i = tid; i < 512; i += 256) {
    float4 vu = u4[i];
    float4 vg = g4[i];
    float4 vd = d4[i];
    int e = i * 4;
    *(float4*)(wb + WU_OFF + (e >> 6) * HSTR + (e & 63)) = vu;
    *(float4*)(wb + WG_OFF + (e >> 6) * HSTR + (e & 63)) = vg;
    *(float4*)(wb + WD_OFF + (e >> 5) * PSTR + (e & 31)) = vd;
  }
  if (tid < 32) wb[BU_OFF + tid] = gbu[(size_t)s * 32 + tid];
}

__global__ __launch_bounds__(WAVES * 32) void
purediv32_wmma_f32(const float* __restrict__ x,
                   const float* __restrict__ gWu,
                   const float* __restrict__ gbu,
                   const float* __restrict__ gWg,
                   const float* __restrict__ gWd,
                   float* __restrict__ out,
                   int nsteps, int ntiles) {
  __shared__ float wbuf[2][WBUF_N];          // double-buffered step weights
  __shared__ float hbuf[WAVES][16 * HSTR];   // per-wave h tile (row-major, padded)
  __shared__ float pbuf[WAVES][16 * PSTR];   // per-wave up*gate scratch

  const int tid  = threadIdx.x;
  const int wave = tid >> 5;
  const int lane = tid & 31;
  const int hi   = lane >> 4;   // half-wave select (K/M offset per WMMA layout)
  const int lrow = lane & 15;
  const int koff = hi * 2;

  int tile = blockIdx.x * WAVES + wave;
  const bool active = tile < ntiles;
  if (!active) tile = 0;        // keep EXEC uniform; WMMA requires full EXEC

  float* hb = hbuf[wave];
  float* pb = pbuf[wave];

  // preload step-0 weights
  load_weights(wbuf[0], gWu, gbu, gWg, gWd, 0, tid);

  // copy the wave's 16x64 slice of x into padded LDS
  {
    const float4* xg = (const float4*)(x + (size_t)tile * 1024);
    #pragma unroll
    for (int i = lane; i < 256; i += 32) {
      float4 v = xg[i];
      int e = i * 4;
      *(float4*)(hb + (e >> 6) * HSTR + (e & 63)) = v;
    }
  }
  __syncthreads();

  // h accumulators in WMMA 16x16 f32 C-layout: acc[d] covers columns 16d..16d+15
  v8f acc[4];
  #pragma unroll
  for (int d = 0; d < 4; ++d)
    #pragma unroll
    for (int r = 0; r < 8; ++r)
      acc[d][r] = hb[(r + 8 * hi) * HSTR + lrow + 16 * d];

  for (int s = 0; s < nsteps; ++s) {
    const float* wb = wbuf[s & 1];
    if (s + 1 < nsteps)
      load_weights(wbuf[(s + 1) & 1], gWu, gbu, gWg, gWd, s + 1, tid);

    // ---- phase 1: up = h @ Wu^T ; gate = h @ Wg^T  (K=64 -> 16 chunks of 4)
    v8f up0 = {}, up1 = {}, ga0 = {}, ga1 = {};
    const float* hr = hb + lrow * HSTR;
    #pragma unroll
    for (int t = 0; t < 16; ++t) {
      v2f a  = *(const v2f*)(hr + 4 * t + koff);                               // A: h[m][k..k+1]
      v2f b0 = *(const v2f*)(wb + WU_OFF + lrow        * HSTR + 4 * t + koff); // B: Wu[n][k..k+1]
      v2f b1 = *(const v2f*)(wb + WU_OFF + (lrow + 16) * HSTR + 4 * t + koff);
      v2f b2 = *(const v2f*)(wb + WG_OFF + lrow        * HSTR + 4 * t + koff);
      v2f b3 = *(const v2f*)(wb + WG_OFF + (lrow + 16) * HSTR + 4 * t + koff);
      up0 = __builtin_amdgcn_wmma_f32_16x16x4_f32(false, a, false, b0, (short)0, up0, false, false);
      up1 = __builtin_amdgcn_wmma_f32_16x16x4_f32(false, a, false, b1, (short)0, up1, false, false);
      ga0 = __builtin_amdgcn_wmma_f32_16x16x4_f32(false, a, false, b2, (short)0, ga0, false, false);
      ga1 = __builtin_amdgcn_wmma_f32_16x16x4_f32(false, a, false, b3, (short)0, ga1, false, false);
    }

    // ---- phase 2: p = relu(up + bu) * gate  (elementwise in C-layout)
    float c0 = wb[BU_OFF + lrow];
    float c1 = wb[BU_OFF + 16 + lrow];
    #pragma unroll
    for (int r = 0; r < 8; ++r) {
      float u0 = fmaxf(up0[r] + c0, 0.0f);
      float u1 = fmaxf(up1[r] + c1, 0.0f);
      pb[(r + 8 * hi) * PSTR + lrow     ] = u0 * ga0[r];
      pb[(r + 8 * hi) * PSTR + lrow + 16] = u1 * ga1[r];
    }
    // cross-lane LDS RAW inside this wave: force DS completion
    asm volatile("s_wait_dscnt 0" ::: "memory");

    // ---- phase 3: h += p @ Wd^T  (K=32 -> 8 chunks; accumulate into h regs)
    #pragma unroll
    for (int t = 0; t < 8; ++t) {
      v2f a = *(const v2f*)(pb + lrow * PSTR + 4 * t + koff);
      #pragma unroll
      for (int d = 0; d < 4; ++d) {
        v2f b = *(const v2f*)(wb + WD_OFF + (lrow + 16 * d) * PSTR + 4 * t + koff);
        acc[d] = __builtin_amdgcn_wmma_f32_16x16x4_f32(false, a, false, b, (short)0, acc[d], false, false);
      }
    }

    // mirror updated h into LDS for next step's A-layout reads
    #pragma unroll
    for (int d = 0; d < 4; ++d)
      #pragma unroll
      for (int r = 0; r < 8; ++r)
        hb[(r + 8 * hi) * HSTR + lrow + 16 * d] = acc[d][r];

    __syncthreads();  // weight buffer handoff + h visibility
  }

  if (active) {
    float* og = out + (size_t)tile * 1024;
    #pragma unroll
    for (int d = 0; d < 4; ++d)
      #pragma unroll
      for (int r = 0; r < 8; ++r)
        og[(r + 8 * hi) * 64 + lrow + 16 * d] = acc[d][r];
  }
}

extern "C" void kernel_launch(void* const* d_in, const int* in_sizes, int n_in,
                              void* d_out, int out_size, void* d_ws, size_t ws_size,
                              hipStream_t stream) {
  const float* x   = (const float*)d_in[0];  // [B, 64]
  const float* Wu  = (const float*)d_in[1];  // [97, 32, 64]
  const float* bu  = (const float*)d_in[2];  // [97, 32]
  const float* Wg  = (const float*)d_in[3];  // [97, 32, 64]
  const float* Wd  = (const float*)d_in[4];  // [97, 64, 32]
  float*       out = (float*)d_out;          // [B, 64]

  const int Btot   = in_sizes[0] / 64;
  const int nsteps = in_sizes[2] / 32;       // 97
  const int ntiles = Btot / 16;              // 4096
  const int blocks = (ntiles + WAVES - 1) / WAVES;

  purediv32_wmma_f32<<<blocks, WAVES * 32, 0, stream>>>(x, Wu, bu, Wg, Wd, out,
                                                        nsteps, ntiles);
}